// JacobiMachine_50964081935582
// MI455X (gfx1250) — compile-verified
//
#include <hip/hip_runtime.h>
#include <hip/hip_bf16.h>
#include <stdint.h>

// Problem constants (match reference)
#define N   2048
#define NT  100
#define F   10                 // fused Jacobi steps per launch (NT % F == 0)

// Tiling: 256 threads/block (8 wave32) produce a 32x128 output tile.
#define TX  128
#define TY  32
#define HA  (TY + 2 * F)       // 52 frame rows (x_0 tile incl. halo)
#define W   (TX + 2 * F + 4)   // 152: LDS row stride; data cols 2..149
// Frame mapping: global (gy,gx) <-> frame (fy,fc): gy = r0-F+fy, gx = c0-F-2+fc.
// The +2 column shift makes the final 32x128 tile quad-aligned (col 12..139).

typedef uint32_t u32x4 __attribute__((ext_vector_type(4)));
typedef uint32_t u32x8 __attribute__((ext_vector_type(8)));

// ---------------------------------------------------------------------------
// Init: x0 = exp(-50*((X-0.5)^2 + (Y-0.5)^2))   (full field incl. boundary)
// ---------------------------------------------------------------------------
__global__ __launch_bounds__(256) void jacobi_init(const float* __restrict__ X,
                                                   const float* __restrict__ Y,
                                                   float* __restrict__ out) {
    int i = blockIdx.x * 256 + threadIdx.x;
    float dx = X[i] - 0.5f;
    float dy = Y[i] - 0.5f;
    out[i] = __expf(-50.0f * (dx * dx + dy * dy));
}

// ---------------------------------------------------------------------------
// One in-LDS Jacobi step s (1..F-1): Q = J(P) on frame rows [s, 51-s],
// quad-columns covering data cols [s+2, 149-s]. Extra quad elements land on
// ring < s cells (garbage-tolerated by the ring invariant).
// ---------------------------------------------------------------------------
template <bool EDGE>
__device__ __forceinline__ void lds_step(const float* P, float* Q,
                                         int s, int tid, int r0, int c0) {
    const int rlo = s;
    const int NR  = HA - 2 * s;                 // rows s .. 51-s
    const int qlo = (s + 2) >> 2;
    const int NQ  = ((149 - s) >> 2) - qlo + 1; // aligned quads covering region
    for (int j = tid; j < NR * NQ; j += 256) {
        const int jr = j / NQ;                  // NQ is a constant per unroll
        const int fy = rlo + jr;
        const int q  = qlo + (j - jr * NQ);
        const int base = fy * W + 4 * q;        // 16B-aligned (W % 4 == 0)
        const float4 up = *(const float4*)(P + base - W);
        const float4 dn = *(const float4*)(P + base + W);
        const float4 ce = *(const float4*)(P + base);
        const float  lf = P[base - 1];
        const float  rt = P[base + 4];
        float4 r;
        r.x = 0.25f * ((up.x + dn.x) + (lf   + ce.y));
        r.y = 0.25f * ((up.y + dn.y) + (ce.x + ce.z));
        r.z = 0.25f * ((up.z + dn.z) + (ce.y + ce.w));
        r.w = 0.25f * ((up.w + dn.w) + (ce.z + rt));
        if (EDGE) {  // Dirichlet mask + zero padding, only for boundary blocks
            const int gy = r0 - F + fy;
            const int gx = c0 - F - 2 + 4 * q;
            const bool zy = (gy < 1) | (gy > N - 2);
            r.x = (zy | (gx     < 1) | (gx     > N - 2)) ? 0.0f : r.x;
            r.y = (zy | (gx + 1 < 1) | (gx + 1 > N - 2)) ? 0.0f : r.y;
            r.z = (zy | (gx + 2 < 1) | (gx + 2 > N - 2)) ? 0.0f : r.z;
            r.w = (zy | (gx + 3 < 1) | (gx + 3 > N - 2)) ? 0.0f : r.w;
        }
        *(float4*)(Q + base) = r;
    }
}

// ---------------------------------------------------------------------------
// Body: F-1 in-LDS steps + final step with coalesced b128 global stores.
// ---------------------------------------------------------------------------
template <bool EDGE>
__device__ __forceinline__ void sweep_body(float* A, float* B,
                                           float* __restrict__ dst,
                                           int tid, int r0, int c0) {
    float* bufs[2] = { A, B };                  // x_s lives in bufs[s & 1]
#pragma unroll
    for (int s = 1; s <= F - 1; ++s) {
        lds_step<EDGE>(bufs[(s - 1) & 1], bufs[s & 1], s, tid, r0, c0);
        __syncthreads();
    }

    // Step F: rows fy = 10..41, quads q = 3..34 (data cols 12..139, exact).
    const float* P = bufs[(F - 1) & 1];         // x_{F-1}
    const int q  = 3 + (tid & 31);
    const int rg = (int)(tid >> 5);             // 0..7
#pragma unroll
    for (int rr = 0; rr < 4; ++rr) {
        const int ty = rg * 4 + rr;             // 0..31
        const int fy = F + ty;
        const int base = fy * W + 4 * q;
        const float4 up = *(const float4*)(P + base - W);
        const float4 dn = *(const float4*)(P + base + W);
        const float4 ce = *(const float4*)(P + base);
        const float  lf = P[base - 1];
        const float  rt = P[base + 4];
        float4 r;
        r.x = 0.25f * ((up.x + dn.x) + (lf   + ce.y));
        r.y = 0.25f * ((up.y + dn.y) + (ce.x + ce.z));
        r.z = 0.25f * ((up.z + dn.z) + (ce.y + ce.w));
        r.w = 0.25f * ((up.w + dn.w) + (ce.z + rt));
        const int gy = r0 + ty;
        const int gx = c0 + 4 * (q - 3);
        if (EDGE) {
            const bool zy = (gy == 0) | (gy == N - 1);
            r.x = (zy | (gx     == 0) | (gx     == N - 1)) ? 0.0f : r.x;
            r.y = (zy | (gx + 1 == 0) | (gx + 1 == N - 1)) ? 0.0f : r.y;
            r.z = (zy | (gx + 2 == 0) | (gx + 2 == N - 1)) ? 0.0f : r.z;
            r.w = (zy | (gx + 3 == 0) | (gx + 3 == N - 1)) ? 0.0f : r.w;
        }
        *(float4*)(dst + (size_t)gy * N + gx) = r;
    }
}

// ---------------------------------------------------------------------------
// Ten fused Jacobi sweeps: dst = J^10(src). One TDM DMA loads the halo-10
// tile into the common LDS frame; 10 sweeps run entirely in LDS.
// ---------------------------------------------------------------------------
__global__ __launch_bounds__(256) void jacobi_sweep10(const float* __restrict__ src,
                                                      float* __restrict__ dst) {
    __shared__ __align__(16) float A[HA * W + 16];   // +slack for TDM row spill
    __shared__ __align__(16) float B[HA * W + 16];

    const int tid = (int)threadIdx.x;
    const int c0 = blockIdx.x * TX;
    const int r0 = blockIdx.y * TY;
    const int cAx = (c0 >= F) ? (c0 - F) : 0;        // clamped load corner
    const int cAy = (r0 >= F) ? (r0 - F) : 0;
    const int sx  = cAx - (c0 - F);                  // 0..F top/left shift
    const int sy  = cAy - (r0 - F);

    if (tid < 32) {  // wave 0 only: TDM ignores EXEC, issue once per block
        const uint64_t ga = (uint64_t)(uintptr_t)src
                          + 4ull * ((uint64_t)cAy * N + (uint64_t)cAx);
        const uint32_t lds_off = (uint32_t)(uintptr_t)A
                               + 4u * (uint32_t)(sy * W + sx + 2);
        const uint32_t td0 = (uint32_t)(N - cAx);    // right OOB -> zero-fill
        const uint32_t td1 = (uint32_t)(N - cAy);    // bottom OOB -> zero-fill
        const uint32_t rows = (uint32_t)(HA - sy);   // rows loaded

        // D# group 0: [1:0]=count=1, [63:32]=lds_addr, [120:64]=global_addr,
        //             [127:126]=type=2
        u32x4 g0;
        g0[0] = 1u;
        g0[1] = lds_off;
        g0[2] = (uint32_t)ga;
        g0[3] = ((uint32_t)(ga >> 32) & 0x01FFFFFFu) | 0x80000000u;

        // D# group 1: data_size=2 (4B), tensor dims, tile W x rows, stride N.
        // tile_dim0 == LDS stride W: row r lands at lds_off + r*W floats.
        u32x8 g1;
        g1[0] = 2u << 16;
        g1[1] = (td0 & 0xFFFFu) << 16;
        g1[2] = (td0 >> 16) | ((td1 & 0xFFFFu) << 16);
        g1[3] = (td1 >> 16) | ((uint32_t)W << 16);
        g1[4] = rows;
        g1[5] = (uint32_t)N;
        g1[6] = 0u;
        g1[7] = 0u;

        asm volatile("tensor_load_to_lds %0, %1"
                     :: "s"(g0), "s"(g1)
                     : "memory");
#if __has_builtin(__builtin_amdgcn_s_wait_tensorcnt)
        __builtin_amdgcn_s_wait_tensorcnt(0);
#else
        asm volatile("s_wait_tensorcnt 0x0" ::: "memory");
#endif
    }
    __syncthreads();

    const bool edge = (r0 == 0) | (c0 == 0) | (r0 == N - TY) | (c0 == N - TX);
    if (edge) sweep_body<true>(A, B, dst, tid, r0, c0);
    else      sweep_body<false>(A, B, dst, tid, r0, c0);
}

// ---------------------------------------------------------------------------
// Launch: init -> d_out, then NT/F fused sweeps ping-ponging d_out <-> d_ws.
// NT/F = 10 is even, so the last fused sweep writes into d_out.
// ---------------------------------------------------------------------------
extern "C" void kernel_launch(void* const* d_in, const int* in_sizes, int n_in,
                              void* d_out, int out_size, void* d_ws, size_t ws_size,
                              hipStream_t stream) {
    (void)in_sizes; (void)n_in; (void)out_size; (void)ws_size;
    const float* X = (const float*)d_in[0];
    const float* Y = (const float*)d_in[1];
    float* out = (float*)d_out;
    float* ws  = (float*)d_ws;          // needs N*N*4 = 16.8 MB of scratch

    jacobi_init<<<dim3((N * N) / 256), dim3(256), 0, stream>>>(X, Y, out);

    float* bufs[2] = { out, ws };
    dim3 grid(N / TX, N / TY);          // 16 x 64 blocks
    dim3 block(256);
    for (int k = 1; k <= NT / F; ++k) { // 10 launches of 10 fused steps
        const float* s = bufs[(k - 1) & 1];
        float*       d = bufs[k & 1];
        jacobi_sweep10<<<grid, block, 0, stream>>>(s, d);
    }
}